// Tight_CLoss_47648367182237
// MI455X (gfx1250) — compile-verified
//
#include <hip/hip_runtime.h>
#include <stdint.h>

#define THRESHOLD 1000.0f
#define B_ROWS 8192
#define C_COLS 32000
#define NF4    (C_COLS / 4)          /* 8000 float4 per row */
#define TPB    256
#define CHUNK_F4 1024                /* float4 per chunk = TPB*4 (16 KB) */
#define NCHUNK ((NF4 + CHUNK_F4 - 1) / CHUNK_F4)   /* 8 */
#define L2E 1.4426950408889634f
#define LN2 0.6931471805599453f

// ---------------------------------------------------------------------------
// Kernel 1: per-row hinge/logsumexp loss, streamed via async global->LDS DMA.
// One 256-thread block per row; 2x16KB LDS double buffer.
// ---------------------------------------------------------------------------
__global__ __launch_bounds__(TPB) void row_loss_kernel(
    const float* __restrict__ logits,
    const int*   __restrict__ target,
    float*       __restrict__ lvec)
{
    __shared__ float4 buf[2][CHUNK_F4];            // 32 KB staging
    __shared__ float r_mm[TPB], r_m[TPB], r_s[TPB], r_lt[TPB];

    const int row = blockIdx.x;
    const int t   = threadIdx.x;
    const int tg  = target[row];
    const uint64_t gbase = (uint64_t)(uintptr_t)(logits + (size_t)row * C_COLS);

    // Issue one chunk (4 async b128 per wave; OOB lanes clamp the global
    // address so every wave's ASYNCcnt increments by exactly 4).
    auto issue = [&](int c, int bsel) {
#pragma unroll
        for (int j = 0; j < 4; ++j) {
            int slot = j * TPB + t;
            int f4   = c * CHUNK_F4 + slot;
            if (f4 > NF4 - 1) f4 = NF4 - 1;        // clamp (data ignored later)
            uint32_t voff = (uint32_t)f4 * 16u;    // byte offset from row base
            uint32_t lds  = (uint32_t)(uintptr_t)(&buf[bsel][slot]);
            asm volatile("global_load_async_to_lds_b128 %0, %1, %2"
                         :: "v"(lds), "v"(voff), "s"(gbase)
                         : "memory");
        }
    };

    float mmax = -__builtin_inff();   // max over columns != target
    float lt   = -__builtin_inff();   // logit at target column
    float m    = -__builtin_inff();   // online logsumexp running max
    float s    = 0.0f;                // online sum of 2^((x-m)*log2e)

    issue(0, 0);
    for (int c = 0; c < NCHUNK; ++c) {
        if (c + 1 < NCHUNK) {
            issue(c + 1, (c + 1) & 1);
            asm volatile("s_wait_asynccnt 0x4" ::: "memory"); // chunk c landed
        } else {
            asm volatile("s_wait_asynccnt 0x0" ::: "memory");
        }
        __syncthreads();   // all waves' chunk-c data visible in LDS

        int valid = NF4 - c * CHUNK_F4;
        if (valid > CHUNK_F4) valid = CHUNK_F4;

#pragma unroll
        for (int j = 0; j < 4; ++j) {
            int slot = j * TPB + t;
            if (slot < valid) {
                float4 v = buf[c & 1][slot];
                int col  = (c * CHUNK_F4 + slot) * 4;
                float xs[4] = {v.x, v.y, v.z, v.w};
#pragma unroll
                for (int k = 0; k < 4; ++k) {
                    float x = xs[k];
                    if (col + k == tg) { lt = x; }
                    else               { mmax = fmaxf(mmax, x); }
                    if (x > m) {                       // rare after warmup
                        s *= __builtin_exp2f((m - x) * L2E);
                        m  = x;
                    }
                    s += __builtin_exp2f((x - m) * L2E);
                }
            }
        }
        __syncthreads();   // buffer may be overwritten next iteration
    }

    // Block-wide reduction: max, target-logit (max works: others hold -inf),
    // and online-softmax state merge.
    r_mm[t] = mmax; r_m[t] = m; r_s[t] = s; r_lt[t] = lt;
    __syncthreads();
    for (int off = TPB / 2; off > 0; off >>= 1) {
        if (t < off) {
            r_mm[t] = fmaxf(r_mm[t], r_mm[t + off]);
            r_lt[t] = fmaxf(r_lt[t], r_lt[t + off]);
            float m1 = r_m[t], s1 = r_s[t];
            float m2 = r_m[t + off], s2 = r_s[t + off];
            float nm = fmaxf(m1, m2);
            r_s[t] = s1 * __builtin_exp2f((m1 - nm) * L2E)
                   + s2 * __builtin_exp2f((m2 - nm) * L2E);
            r_m[t] = nm;
        }
        __syncthreads();
    }
    if (t == 0) {
        float logit_t = r_lt[0];
        float margin  = logit_t - r_mm[0];
        float lse     = r_m[0] + LN2 * __builtin_log2f(r_s[0]);
        float shl = (margin >= 0.0f) ? (1.0f - margin)
                                     : (1.0f - logit_t + lse);
        lvec[row] = fmaxf(shl, 0.0f);
    }
}

// ---------------------------------------------------------------------------
// Kernel 2: sort 8192 losses, cumsum, keep-mask, final scalar. One block.
// ---------------------------------------------------------------------------
#define TPB2 1024
#define EPT  (B_ROWS / TPB2)   /* 8 elements per thread */

__global__ __launch_bounds__(TPB2) void curriculum_kernel(
    const float* __restrict__ lvec,
    float*       __restrict__ outv)
{
    __shared__ float A[B_ROWS];     // 32 KB: losses, then sorted in place
    __shared__ float sums[TPB2];    // block-sum scan
    __shared__ float rs[TPB2], rc[TPB2];

    const int t = threadIdx.x;
    for (int i = t; i < B_ROWS; i += TPB2) A[i] = lvec[i];
    __syncthreads();

    // Bitonic sort, ascending. Pairs within a (k,j) stage are disjoint.
    for (int k = 2; k <= B_ROWS; k <<= 1) {
        for (int j = k >> 1; j > 0; j >>= 1) {
            for (int i = t; i < B_ROWS; i += TPB2) {
                int ixj = i ^ j;
                if (ixj > i) {
                    bool up = ((i & k) == 0);
                    float a = A[i], b = A[ixj];
                    if (up ? (a > b) : (a < b)) { A[i] = b; A[ixj] = a; }
                }
            }
            __syncthreads();
        }
    }

    // Blocked inclusive scan: serial scan of 8 owned elements in registers,
    // then Hillis-Steele over the 1024 per-thread totals.
    float v[EPT], ps[EPT];
    const int base = t * EPT;
    float run = 0.0f;
#pragma unroll
    for (int k = 0; k < EPT; ++k) { v[k] = A[base + k]; run += v[k]; ps[k] = run; }
    sums[t] = run;
    __syncthreads();
    for (int off = 1; off < TPB2; off <<= 1) {
        float x = sums[t] + ((t >= off) ? sums[t - off] : 0.0f);
        __syncthreads();
        sums[t] = x;
        __syncthreads();
    }
    const float offset = (t > 0) ? sums[t - 1] : 0.0f;

    // keep_i = (csum_i <= THRESHOLD + 1 - i); accumulate kept sum / count.
    float ksum = 0.0f, kcnt = 0.0f;
#pragma unroll
    for (int k = 0; k < EPT; ++k) {
        float csum = ps[k] + offset;
        int   i    = base + k;
        if (csum <= (THRESHOLD + 1.0f - (float)i)) { ksum += v[k]; kcnt += 1.0f; }
    }
    rs[t] = ksum; rc[t] = kcnt;
    __syncthreads();
    for (int off = TPB2 / 2; off > 0; off >>= 1) {
        if (t < off) { rs[t] += rs[t + off]; rc[t] += rc[t + off]; }
        __syncthreads();
    }
    if (t == 0) {
        float l1 = rs[0];
        float l2 = (float)B_ROWS - rc[0];
        outv[0] = (l1 < l2) ? l2 : l1;
    }
}

// ---------------------------------------------------------------------------
extern "C" void kernel_launch(void* const* d_in, const int* in_sizes, int n_in,
                              void* d_out, int out_size, void* d_ws, size_t ws_size,
                              hipStream_t stream)
{
    const float* logits = (const float*)d_in[0];   // (8192, 32000) fp32
    const int*   target = (const int*)d_in[1];     // (8192,) int32
    float* lvec = (float*)d_ws;                    // 8192 per-row losses

    row_loss_kernel<<<B_ROWS, TPB, 0, stream>>>(logits, target, lvec);
    curriculum_kernel<<<1, TPB2, 0, stream>>>(lvec, (float*)d_out);
}